// Laplacian3DLayer_14998025798278
// MI455X (gfx1250) — compile-verified
//
#include <hip/hip_runtime.h>
#include <cstdint>

// 3-D Laplacian (zero-padded 7-point stencil) on (8,1,192,192,192) f32.
// Memory-bound: ~0.5 GB moved -> ~20 us floor at 23.3 TB/s on MI455X.
// Uses the CDNA5 async global->LDS path (GLOBAL_LOAD_ASYNC_TO_LDS_B32,
// tracked by ASYNCcnt / s_wait_asynccnt) with a 4-plane LDS ring buffer
// so the DMA of plane z+2 overlaps the stencil compute of plane z.

#define NB   8          // batches
#define NN   192        // cube edge
#define TX   30         // interior tile x
#define TY   14         // interior tile y
#define LX   32         // TX + 2 halo
#define LYH  16         // TY + 2 halo
#define PLANE (LX * LYH) // 512 floats per staged plane
#define NTHREADS 256     // 8 wave32s; 512/256 = exactly 2 load slots/thread

// Stage one z-plane (zp) of the tile (with halo) into ring buffer `nb`.
// Valid elements are issued as async DMA into LDS; out-of-range halo slots
// (x/y outside [0,192) or zp outside [0,192)) are ds-stored as 0.0f to
// implement the zero padding of the reference.
__device__ __forceinline__ void stage_plane(const float* __restrict__ x,
                                            float (*buf)[PLANE], int nb,
                                            int tid, int b, int zp,
                                            int x0, int y0) {
  for (int s = tid; s < PLANE; s += NTHREADS) {   // exactly 2 iterations
    const int lx = s & (LX - 1);
    const int ly = s >> 5;
    const int gx = x0 + lx - 1;
    const int gy = y0 + ly - 1;
    const bool valid = (zp >= 0) && (zp < NN) &&
                       ((unsigned)gx < (unsigned)NN) &&
                       ((unsigned)gy < (unsigned)NN);
    if (valid) {
      // byte offset from base pointer; max 8*192^3*4 = 226 MB < 2^31
      const uint32_t goff =
          (uint32_t)((((b * NN + zp) * NN + gy) * NN + gx) * 4);
      // low 32 bits of the shared-aperture flat address == LDS byte offset
      const uint32_t laddr = (uint32_t)(uintptr_t)(&buf[nb][s]);
      asm volatile("global_load_async_to_lds_b32 %0, %1, %2"
                   :: "v"(laddr), "v"(goff), "s"(x)
                   : "memory");
    } else {
      buf[nb][s] = 0.0f;   // zero padding
    }
  }
}

__global__ void __launch_bounds__(NTHREADS)
laplacian3d_async_kernel(const float* __restrict__ x, float* __restrict__ out) {
  __shared__ float buf[4][PLANE];   // 4-plane ring, 8 KB of 320 KB WGP LDS

  const int tid = (int)threadIdx.x;
  const int x0  = (int)blockIdx.x * TX;
  const int y0  = (int)blockIdx.y * TY;
  const int b   = (int)blockIdx.z;

  // Prologue: plane z=-1 is all zeros (ring slot 3); planes 0 and 1 async.
  for (int s = tid; s < PLANE; s += NTHREADS) buf[3][s] = 0.0f;
  stage_plane(x, buf, 0, tid, b, 0, x0, y0);
  stage_plane(x, buf, 1, tid, b, 1, x0, y0);

  const int lx  = tid & (LX - 1);
  const int ly0 = tid >> 5;        // 0..7; each thread covers ly0 and ly0+8

  for (int z = 0; z < NN; ++z) {
    // Drain our wave's async DMAs (plane z+1 issued last iteration), then
    // barrier so every wave's staged data is visible in LDS.
    asm volatile("s_wait_asynccnt 0" ::: "memory");
    __syncthreads();

    // Prefetch plane z+2 into the slot that held plane z-2 (not read this
    // iteration) -> DMA overlaps the compute below; waited next iteration.
    stage_plane(x, buf, (z + 2) & 3, tid, b, z + 2, x0, y0);

    const int pm = (z + 3) & 3;    // plane z-1
    const int pc =  z      & 3;    // plane z
    const int pp = (z + 1) & 3;    // plane z+1

#pragma unroll
    for (int ly = ly0; ly < LYH; ly += 8) {
      if (lx >= 1 && lx <= TX && ly >= 1 && ly <= TY) {
        const int gx = x0 + lx - 1;
        const int gy = y0 + ly - 1;
        if (gx < NN && gy < NN) {
          const int s = (ly << 5) + lx;
          const float c = buf[pc][s];
          const float v = buf[pm][s] + buf[pp][s]
                        + buf[pc][s - 1]  + buf[pc][s + 1]
                        + buf[pc][s - LX] + buf[pc][s + LX]
                        - 6.0f * c;
          out[(size_t)(((b * NN + z) * NN + gy) * NN + gx)] = v;
        }
      }
    }
    __syncthreads();  // compute reads done before next staging overwrites
  }
}

extern "C" void kernel_launch(void* const* d_in, const int* in_sizes, int n_in,
                              void* d_out, int out_size, void* d_ws, size_t ws_size,
                              hipStream_t stream) {
  (void)in_sizes; (void)n_in; (void)out_size; (void)d_ws; (void)ws_size;
  const float* x = (const float*)d_in[0];
  float* out = (float*)d_out;
  dim3 grid((NN + TX - 1) / TX,   // 7
            (NN + TY - 1) / TY,   // 14
            NB);                  // 8
  laplacian3d_async_kernel<<<grid, NTHREADS, 0, stream>>>(x, out);
}